// GCNLayer_44908178047711
// MI455X (gfx1250) — compile-verified
//
#include <hip/hip_runtime.h>
#include <cstdint>

#define F 128
#define EPSV 1e-5f
#define NSLOPE 0.01f

typedef float v2f __attribute__((ext_vector_type(2)));
typedef float v8f __attribute__((ext_vector_type(8)));

// ---------------- init: deg=1.0 (self loop), stats=0 ----------------
__global__ void __launch_bounds__(256) k_init(float* deg, float* stats, int N) {
    int i = blockIdx.x * blockDim.x + threadIdx.x;
    if (i < N) deg[i] = 1.0f;
    if (i < 2 * F) stats[i] = 0.0f;
}

// ---------------- degree: atomic count of dst occurrences ----------------
__global__ void __launch_bounds__(256) k_degree(const int* __restrict__ ei, float* deg, int E) {
    int e = blockIdx.x * blockDim.x + threadIdx.x;
    if (e < E) unsafeAtomicAdd(&deg[ei[E + e]], 1.0f);
}

// ---------------- deg -> deg^-1/2 in place (deg >= 1 always) ----------------
__global__ void __launch_bounds__(256) k_rsqrt(float* deg, int N) {
    int i = blockIdx.x * blockDim.x + threadIdx.x;
    if (i < N) deg[i] = rsqrtf(deg[i]);
}

// ---------------- h = x @ W via V_WMMA_F32_16X16X4_F32 ----------------
// 8 waves/block; each wave: one 16-row tile x all 8 column tiles.
// W staged in LDS as K-pair-interleaved float2 so each B fragment is one
// aligned ds_load_b64 straight into a contiguous even VGPR pair (no shuffles).
__global__ void __launch_bounds__(256) k_gemm(const float* __restrict__ x,
                                              const float* __restrict__ W,
                                              float* __restrict__ h, int nTiles) {
    __shared__ v2f Wp[(F / 2) * F];  // 64 KB: Wp[kp*F+col] = (W[2kp][col], W[2kp+1][col])
    for (int idx = threadIdx.x; idx < (F / 2) * F; idx += 256) {
        int kp = idx >> 7, col = idx & (F - 1);
        v2f v;
        v.x = W[(2 * kp) * F + col];
        v.y = W[(2 * kp + 1) * F + col];
        Wp[idx] = v;
    }
    __syncthreads();

    const int wave = threadIdx.x >> 5;
    const int lane = threadIdx.x & 31;
    const int tile = blockIdx.x * 8 + wave;
    if (tile >= nTiles) return;

    const int row0  = tile * 16;
    const int mrow  = row0 + (lane & 15);
    const int hi    = lane >> 4;   // lanes 0-15: K=0,1 ; lanes 16-31: K=2,3
    const int khalf = hi * 2;

    v8f acc[8] = {};
    for (int k = 0; k < F; k += 4) {
        // A fragment: 16x4 f32, row mrow, cols k+khalf .. k+khalf+1 (8B aligned)
        const v2f a = *(const v2f*)(x + (size_t)mrow * F + k + khalf);
        // B fragments: pair (K=k+khalf, K=k+khalf+1) at col; one b64 per tile,
        // immediate offsets c*16*sizeof(v2f) = c*128 bytes off one base.
        const v2f* wrow = Wp + ((k >> 1) + hi) * F + (lane & 15);
#pragma unroll
        for (int c = 0; c < 8; ++c) {
            const v2f bb = wrow[c * 16];
            acc[c] = __builtin_amdgcn_wmma_f32_16x16x4_f32(
                false, a, false, bb, (short)0, acc[c], false, false);
        }
    }

    // C/D layout: VGPR r -> (lanes 0-15: M=r, N=lane) (lanes 16-31: M=r+8, N=lane-16)
    const int rbase = row0 + (hi ? 8 : 0);
#pragma unroll
    for (int c = 0; c < 8; ++c) {
        const int col = c * 16 + (lane & 15);
#pragma unroll
        for (int r = 0; r < 8; ++r)
            h[(size_t)(rbase + r) * F + col] = acc[c][r];
    }
}

// scalar tail for rows not covered by full 16-row tiles (no-op when N%16==0)
__global__ void __launch_bounds__(256) k_gemm_tail(const float* __restrict__ x,
                                                   const float* __restrict__ W,
                                                   float* __restrict__ h,
                                                   int rowStart, int N) {
    int idx = blockIdx.x * blockDim.x + threadIdx.x;
    int row = rowStart + idx / F, col = idx % F;
    if (row >= N) return;
    float s = 0.f;
    for (int k = 0; k < F; ++k) s += x[(size_t)row * F + k] * W[k * F + col];
    h[(size_t)row * F + col] = s;
}

// ---------------- agg init: bias + self-loop contribution ----------------
__global__ void __launch_bounds__(256) k_init_agg(const float* __restrict__ h,
                                                  const float* __restrict__ dis,
                                                  const float* __restrict__ b,
                                                  float* __restrict__ agg, int N) {
    size_t i = (size_t)blockIdx.x * blockDim.x + threadIdx.x;
    if (i >= (size_t)N * F) return;
    int n = (int)(i >> 7), f = (int)(i & (F - 1));
    float w = dis[n];
    agg[i] = b[f] + h[i] * w * w;  // self loop norm = 1/deg
}

// ---------------- SpMM scatter: one wave per edge, 4 features/lane ----------------
__global__ void __launch_bounds__(256) k_spmm(const int* __restrict__ ei,
                                              const float* __restrict__ h,
                                              const float* __restrict__ dis,
                                              float* __restrict__ agg, int E) {
    int wv = (int)(((size_t)blockIdx.x * blockDim.x + threadIdx.x) >> 5);
    int lane = threadIdx.x & 31;
    if (wv >= E) return;
    int s = ei[wv], d = ei[E + wv];
    float norm = dis[s] * dis[d];
    const float* hp = h + (size_t)s * F;
    float* ap = agg + (size_t)d * F;
#pragma unroll
    for (int j = 0; j < 4; ++j) {
        int f = lane + j * 32;  // coalesced across the wave
        unsafeAtomicAdd(ap + f, hp[f] * norm);
    }
}

// ---------------- per-feature sum and sum-of-squares ----------------
__global__ void __launch_bounds__(F) k_stats(const float* __restrict__ agg,
                                             float* __restrict__ stats, int N) {
    int f = threadIdx.x;
    float s1 = 0.f, s2 = 0.f;
    for (int n = blockIdx.x; n < N; n += gridDim.x) {
        float v = agg[(size_t)n * F + f];  // coalesced row reads
        s1 += v; s2 += v * v;
    }
    unsafeAtomicAdd(&stats[f], s1);
    unsafeAtomicAdd(&stats[F + f], s2);
}

// ---------------- fold GraphNorm into per-feature scale/shift ----------------
__global__ void k_scale_shift(const float* __restrict__ stats,
                              const float* __restrict__ gw,
                              const float* __restrict__ gb,
                              const float* __restrict__ gms,
                              float* __restrict__ ss, float invN) {
    int f = threadIdx.x;
    float mean = stats[f] * invN;
    float m = mean * gms[f];
    float var = stats[F + f] * invN - 2.f * m * mean + m * m;
    float inv = rsqrtf(var + EPSV);
    ss[f] = inv * gw[f];
    ss[F + f] = gb[f] - m * inv * gw[f];
}

// ---------------- normalize + LeakyReLU, in-place on d_out ----------------
__global__ void __launch_bounds__(256) k_norm(float* __restrict__ out,
                                              const float* __restrict__ ss, int N) {
    size_t i = (size_t)blockIdx.x * blockDim.x + threadIdx.x;
    if (i >= (size_t)N * F) return;
    int f = (int)(i & (F - 1));
    float v = out[i] * ss[f] + ss[F + f];
    out[i] = v > 0.f ? v : NSLOPE * v;
}

extern "C" void kernel_launch(void* const* d_in, const int* in_sizes, int n_in,
                              void* d_out, int out_size, void* d_ws, size_t ws_size,
                              hipStream_t stream) {
    const float* x   = (const float*)d_in[0];
    const int*   ei  = (const int*)d_in[1];
    const float* W   = (const float*)d_in[2];
    const float* b   = (const float*)d_in[3];
    const float* gw  = (const float*)d_in[4];
    const float* gb  = (const float*)d_in[5];
    const float* gms = (const float*)d_in[6];

    const int N = in_sizes[0] / F;
    const int E = in_sizes[1] / 2;

    // workspace: h [N*F] | deg/dis [N] | stats [2F] | scale/shift [2F]
    float* h     = (float*)d_ws;
    float* deg   = h + (size_t)N * F;
    float* stats = deg + N;
    float* ss    = stats + 2 * F;
    float* agg   = (float*)d_out;  // accumulate directly in output

    const int nTiles = N / 16;
    const int tailStart = nTiles * 16;
    const int tailRows  = N - tailStart;

    k_init  <<<(N + 255) / 256, 256, 0, stream>>>(deg, stats, N);
    k_degree<<<(E + 255) / 256, 256, 0, stream>>>(ei, deg, E);
    k_rsqrt <<<(N + 255) / 256, 256, 0, stream>>>(deg, N);

    k_gemm<<<(nTiles + 7) / 8, 256, 0, stream>>>(x, W, h, nTiles);
    if (tailRows > 0)
        k_gemm_tail<<<(tailRows * F + 255) / 256, 256, 0, stream>>>(x, W, h, tailStart, N);

    k_init_agg<<<(int)(((size_t)N * F + 255) / 256), 256, 0, stream>>>(h, deg, b, agg, N);
    k_spmm<<<(E + 7) / 8, 256, 0, stream>>>(ei, h, deg, agg, E);

    k_stats<<<512, F, 0, stream>>>(agg, stats, N);
    k_scale_shift<<<1, F, 0, stream>>>(stats, gw, gb, gms, ss, 1.0f / (float)N);
    k_norm<<<(int)(((size_t)N * F + 255) / 256), 256, 0, stream>>>(agg, ss, N);

    // tuple output: append edge_index verbatim after the [N,F] float block
    hipMemcpyAsync((char*)d_out + (size_t)N * F * sizeof(float), d_in[1],
                   (size_t)in_sizes[1] * sizeof(int), hipMemcpyDeviceToDevice, stream);
}